// GCNv2_extractKP_43602507989181
// MI455X (gfx1250) — compile-verified
//
#include <hip/hip_runtime.h>

// B=16, K=64, H=128, W=128 ; out = [keypoint(16,64,2), tf_keypoint(16,64,2)] f32
typedef __attribute__((ext_vector_type(2))) float v2f;
typedef __attribute__((ext_vector_type(8))) float v8f;

__device__ __forceinline__ float readlane_f(float v, int lane) {
  return __builtin_bit_cast(float,
      __builtin_amdgcn_readlane(__builtin_bit_cast(int, v), lane));
}

__global__ __launch_bounds__(256)
void centroid_wmma_kernel(const float* __restrict__ in0,
                          const float* __restrict__ in1,
                          float* __restrict__ out)
{
  const int t   = blockIdx.x;          // 0..2047
  const int p   = t >> 10;             // which input tensor
  const int rem = t & 1023;            // b*64 + k
  const float* __restrict__ src =
      (p == 0 ? in0 : in1) + (size_t)rem * (128 * 128);

  const int tid  = threadIdx.x;
  const int wave = tid >> 5;           // 8 waves -> 8 row slabs of 16
  const int lane = tid & 31;
  const int hi   = lane >> 4;          // K-half (and M-half for D)
  const int row  = (wave << 4) + (lane & 15);

  // B-matrix lane constants (B is 4x16; column n lives in lanes n and n+16).
  // Column 0 = ones (-> row sums), column 1 = x-coordinates (-> sum j*v).
  const float isOne = (lane == 0 || lane == 16) ? 1.f : 0.f;
  const float isW   = (lane == 1 || lane == 17) ? 1.f : 0.f;
  const float wofs  = (lane == 17) ? 4.f : 0.f;   // lanes>=16 hold K=2,3 slots

  // lane L<16 loads row cols [8j..8j+3]; lane L>=16 loads cols [8j+4..8j+7].
  const float4* srcv = reinterpret_cast<const float4*>(src) + row * 32 + hi;

  v8f c = {};   // D/C accumulator: D[:,0]=Sum v per row, D[:,1]=Sum j*v per row
#pragma unroll
  for (int j = 0; j < 16; ++j) {
    float4 f = srcv[2 * j];

    const float wbase = (float)(8 * j);
    const float w     = isW * (wbase + wofs);   // lane1: 8j ; lane17: 8j+4

    v2f a0; a0[0] = f.x; a0[1] = f.y;           // cols {8j,8j+1 | 8j+4,8j+5}
    v2f a1; a1[0] = f.z; a1[1] = f.w;           // cols {8j+2,8j+3 | 8j+6,8j+7}

    v2f b0; b0[0] = isOne + w;                  // weights matching a0's columns
            b0[1] = isOne + w + isW;
    v2f b1; b1[0] = b0[0] + 2.f * isW;          // +2 shifts weights for a1
            b1[1] = b0[1] + 2.f * isW;

    c = __builtin_amdgcn_wmma_f32_16x16x4_f32(false, a0, false, b0,
                                              (short)0, c, false, false);
    c = __builtin_amdgcn_wmma_f32_16x16x4_f32(false, a1, false, b1,
                                              (short)0, c, false, false);
  }

  // D layout: VGPR r, lanes 0-15 -> M=r, N=lane ; lanes 16-31 -> M=8+r, N=lane-16
  // lane 0/16 : per-row Sum v  ; lane 1/17 : per-row Sum j*v
  const float rbase = (float)((wave << 4) + (hi << 3));
  float localS = 0.f, localSy = 0.f;
#pragma unroll
  for (int r = 0; r < 8; ++r) {
    localS  += c[r];
    localSy += (rbase + (float)r) * c[r];
  }

  const float S  = readlane_f(localS, 0)  + readlane_f(localS, 16);
  const float Sx = readlane_f(localS, 1)  + readlane_f(localS, 17);
  const float Sy = readlane_f(localSy, 0) + readlane_f(localSy, 16);

  __shared__ float acc[8][3];
  if (lane == 0) { acc[wave][0] = S; acc[wave][1] = Sx; acc[wave][2] = Sy; }
  __syncthreads();

  if (tid == 0) {
    float s = 0.f, sx = 0.f, sy = 0.f;
#pragma unroll
    for (int w2 = 0; w2 < 8; ++w2) {
      s  += acc[w2][0];
      sx += acc[w2][1];
      sy += acc[w2][2];
    }
    float* o = out + (size_t)p * 2048 + (size_t)rem * 2;
    o[0] = rintf(sx / s);   // rint = round-half-even, matches jnp.round
    o[1] = rintf(sy / s);
  }
}

extern "C" void kernel_launch(void* const* d_in, const int* in_sizes, int n_in,
                              void* d_out, int out_size, void* d_ws, size_t ws_size,
                              hipStream_t stream) {
  const float* a = (const float*)d_in[0];
  const float* b = (const float*)d_in[1];
  float* out = (float*)d_out;
  hipLaunchKernelGGL(centroid_wmma_kernel, dim3(2048), dim3(256), 0, stream,
                     a, b, out);
}